// GATConvolution_44633300140786
// MI455X (gfx1250) — compile-verified
//
#include <hip/hip_runtime.h>
#include <math.h>

typedef float v2f __attribute__((ext_vector_type(2)));
typedef float v8f __attribute__((ext_vector_type(8)));

#define IN_F    128
#define OUT_F   128
#define N_HEAD  3
#define ATTR_F  24      // NUM_HEADS * QUERY_DIM
#define AW_COLS 48
#define LDS_PAD 4       // row stride 132: bank k+4r per row -> conflict-free
#define ROW_STRIDE (IN_F + LDS_PAD)

// ---------------------------------------------------------------------------
// float atomic max via signed/unsigned integer ordering trick.
// Requires destination initialized to -inf (0xFF800000).
// ---------------------------------------------------------------------------
__device__ __forceinline__ void atomic_max_float(float* addr, float val) {
    if (!(val < 0.0f)) {
        atomicMax(reinterpret_cast<int*>(addr), __float_as_int(val));
    } else {
        atomicMin(reinterpret_cast<unsigned int*>(addr), __float_as_uint(val));
    }
}

// ---------------------------------------------------------------------------
// Kernel 0: init accumulators (harness poisons buffers; atomics need zeros)
// ---------------------------------------------------------------------------
__global__ void gat_init_kernel(float* __restrict__ out, int n_out,
                                float* __restrict__ segmax,
                                float* __restrict__ denom, int n_seg) {
    int i = blockIdx.x * blockDim.x + threadIdx.x;
    if (i < n_out) out[i] = 0.0f;
    if (i < n_seg) {
        segmax[i] = -INFINITY;
        denom[i]  = 0.0f;
    }
}

// ---------------------------------------------------------------------------
// Kernel 1: node projections via V_WMMA_F32_16X16X4_F32.
//   h    = x @ lw^T + lb   [N,128]
//   attr = x @ Wq^T + bq   [N,24]   (2 tiles, second tile column-guarded)
// Block = 128 threads (4 waves), 16 nodes per block.
// X tile staged in LDS (shared); each wave stages its 16-col weight tile in a
// private LDS slab (coalesced b128 loads, zero-filled invalid rows), so the
// inner loop is exactly: ds_load_2addr_b64 x2 + v_wmma chain.
// Rows padded to 132 floats: lane r of a fragment read hits bank (k+4r)%64,
// all 32 lanes distinct -> zero LDS bank conflicts.
// Full-tile fast paths keep EXEC untouched in load & store epilogues.
// ---------------------------------------------------------------------------
__global__ __launch_bounds__(128) void gat_node_proj_kernel(
    const float* __restrict__ x,   // [N,128]
    const float* __restrict__ lw,  // [128,128]
    const float* __restrict__ lb,  // [128]
    const float* __restrict__ Wq,  // [24,128]
    const float* __restrict__ bq,  // [24]
    float* __restrict__ h,         // [N,128]
    float* __restrict__ attr,      // [N,24]
    int n_nodes)
{
    __shared__ float Xs[16][ROW_STRIDE];        // ~8.5 KB, block-shared
    __shared__ float Ws[4][16][ROW_STRIDE];     // ~33.8 KB, one slab per wave

    const int  tid      = threadIdx.x;
    const int  wave     = tid >> 5;
    const int  lane     = tid & 31;
    const int  nodeBase = blockIdx.x * 16;
    const bool fullTile = (nodeBase + 16 <= n_nodes);

    // Prefetch next block's X tile while we work on this one.
    {
        int nextNode = nodeBase + 16;
        if (nextNode + (tid >> 3) < n_nodes)
            __builtin_prefetch(x + ((size_t)(nextNode + (tid >> 3)) * IN_F) +
                                   (tid & 7) * 16, 0, 1);
    }

    // Cooperative load of 16x128 fp32 X tile (512 float4, 4 per thread).
    {
        const float4* xg = reinterpret_cast<const float4*>(x);
        if (fullTile) {
#pragma unroll
            for (int i = 0; i < 4; ++i) {
                int idx  = tid + i * 128;      // 0..511
                int jrow = idx >> 5;
                int jcol = idx & 31;
                *reinterpret_cast<float4*>(&Xs[jrow][jcol * 4]) =
                    xg[(size_t)(nodeBase + jrow) * (IN_F / 4) + jcol];
            }
        } else {
#pragma unroll
            for (int i = 0; i < 4; ++i) {
                int idx  = tid + i * 128;
                int jrow = idx >> 5;
                int jcol = idx & 31;
                int node = nodeBase + jrow;
                float4 v = make_float4(0.f, 0.f, 0.f, 0.f);
                if (node < n_nodes) v = xg[(size_t)node * (IN_F / 4) + jcol];
                *reinterpret_cast<float4*>(&Xs[jrow][jcol * 4]) = v;
            }
        }
    }
    __syncthreads();

    const int row   = lane & 15;             // A-fragment row (M) / B column (N)
    const int khalf = (lane >> 4) * 2;       // K sub-offset for this lane half
    const int mBase = (lane >> 4) * 8;       // C-fragment row base

    // 10 output tiles: t 0..7 -> h (cols t*16..), t 8..9 -> attr.
    for (int t = wave; t < 10; t += 4) {
        const bool   isAttr  = (t >= 8);
        const int    colBase = isAttr ? (t - 8) * 16 : t * 16;
        const int    nCols   = isAttr ? ATTR_F : OUT_F;
        const float* W       = isAttr ? Wq : lw;
        const float* bias    = isAttr ? bq : lb;
        const int    col     = colBase + row;      // this lane's N column
        const bool   colOk   = (col < nCols);

        // Wave-private staging of the 16x128 weight tile (coalesced b128).
        {
            const float4* wg = reinterpret_cast<const float4*>(W);
#pragma unroll
            for (int i = 0; i < 16; ++i) {
                int idx  = lane + i * 32;   // 0..511
                int wr   = idx >> 5;        // weight row within tile
                int wc   = idx & 31;        // float4 column
                int grow = colBase + wr;
                float4 v = make_float4(0.f, 0.f, 0.f, 0.f);
                if (grow < nCols) v = wg[(size_t)grow * (IN_F / 4) + wc];
                *reinterpret_cast<float4*>(&Ws[wave][wr][wc * 4]) = v;
            }
        }
        // Slab is private to this wave: no barrier; compiler waits DScnt.

        const float bval = colOk ? bias[col] : 0.0f;
        v8f c;
#pragma unroll
        for (int i = 0; i < 8; ++i) c[i] = bval;

        const float* xrow = &Xs[row][khalf];
        const float* wrow = &Ws[wave][row][khalf];

#pragma unroll 8
        for (int k = 0; k < IN_F; k += 4) {
            v2f a;
            a[0] = xrow[k];
            a[1] = xrow[k + 1];
            v2f b;
            b[0] = wrow[k];
            b[1] = wrow[k + 1];
            // emits v_wmma_f32_16x16x4_f32
            c = __builtin_amdgcn_wmma_f32_16x16x4_f32(
                    false, a, false, b, (short)0, c, false, false);
        }

        if (colOk) {
            float* dstp = isAttr ? attr : h;
            if (fullTile) {
#pragma unroll
                for (int i = 0; i < 8; ++i)
                    dstp[(size_t)(nodeBase + mBase + i) * nCols + col] = c[i];
            } else {
#pragma unroll
                for (int i = 0; i < 8; ++i) {
                    int node = nodeBase + mBase + i;
                    if (node < n_nodes)
                        dstp[(size_t)node * nCols + col] = c[i];
                }
            }
        }
    }
}

// ---------------------------------------------------------------------------
// Kernel 2: per-edge attention logits + segment max (atomic float max).
//   e[E,3] = concat(attr[s], attr[r]) @ aw^T + ab
// ---------------------------------------------------------------------------
__global__ __launch_bounds__(256) void gat_edge_logits_kernel(
    const long long* __restrict__ src,
    const long long* __restrict__ dst,
    const float* __restrict__ attr,   // [N,24]
    const float* __restrict__ aw,     // [3,48]
    const float* __restrict__ ab,     // [3]
    float* __restrict__ elog,         // [E,3]
    float* __restrict__ segmax,       // [N,3]
    int n_edges)
{
    __shared__ float aws[N_HEAD * AW_COLS];
    __shared__ float abs_[N_HEAD];
    if (threadIdx.x < N_HEAD * AW_COLS) aws[threadIdx.x] = aw[threadIdx.x];
    if (threadIdx.x < N_HEAD)           abs_[threadIdx.x] = ab[threadIdx.x];
    __syncthreads();

    int e = blockIdx.x * blockDim.x + threadIdx.x;
    if (e >= n_edges) return;

    const int s = (int)src[e];
    const int r = (int)dst[e];

    float av[2 * ATTR_F];
    {
        const float4* ps = reinterpret_cast<const float4*>(attr + (size_t)s * ATTR_F);
        const float4* pr = reinterpret_cast<const float4*>(attr + (size_t)r * ATTR_F);
        float4* reg = reinterpret_cast<float4*>(av);
#pragma unroll
        for (int i = 0; i < ATTR_F / 4; ++i) reg[i] = ps[i];
#pragma unroll
        for (int i = 0; i < ATTR_F / 4; ++i) reg[ATTR_F / 4 + i] = pr[i];
    }

    float acc0 = abs_[0], acc1 = abs_[1], acc2 = abs_[2];
#pragma unroll
    for (int i = 0; i < AW_COLS; ++i) {
        float v = av[i];
        acc0 = fmaf(v, aws[0 * AW_COLS + i], acc0);
        acc1 = fmaf(v, aws[1 * AW_COLS + i], acc1);
        acc2 = fmaf(v, aws[2 * AW_COLS + i], acc2);
    }

    elog[(size_t)e * N_HEAD + 0] = acc0;
    elog[(size_t)e * N_HEAD + 1] = acc1;
    elog[(size_t)e * N_HEAD + 2] = acc2;

    atomic_max_float(&segmax[(size_t)r * N_HEAD + 0], acc0);
    atomic_max_float(&segmax[(size_t)r * N_HEAD + 1], acc1);
    atomic_max_float(&segmax[(size_t)r * N_HEAD + 2], acc2);
}

// ---------------------------------------------------------------------------
// Kernel 3: exponentiate (stabilized) + segment-sum denominators.
// ---------------------------------------------------------------------------
__global__ __launch_bounds__(256) void gat_edge_exp_kernel(
    const long long* __restrict__ dst,
    const float* __restrict__ segmax,   // [N,3]
    float* __restrict__ elog,           // [E,3] in: logits, out: exp()
    float* __restrict__ denom,          // [N,3]
    int n_edges)
{
    int e = blockIdx.x * blockDim.x + threadIdx.x;
    if (e >= n_edges) return;
    const int r = (int)dst[e];
#pragma unroll
    for (int hh = 0; hh < N_HEAD; ++hh) {
        float v = __expf(elog[(size_t)e * N_HEAD + hh] -
                         segmax[(size_t)r * N_HEAD + hh]);
        elog[(size_t)e * N_HEAD + hh] = v;
        atomicAdd(&denom[(size_t)r * N_HEAD + hh], v);
    }
}

// ---------------------------------------------------------------------------
// Kernel 4: attention-weighted scatter. One wave32 per edge: each lane moves
// a float4 of h[s] scaled by mean-head alpha, atomic-adds into out[r].
// ---------------------------------------------------------------------------
__global__ __launch_bounds__(256) void gat_edge_aggregate_kernel(
    const long long* __restrict__ src,
    const long long* __restrict__ dst,
    const float* __restrict__ hproj,   // [N,128]
    const float* __restrict__ eexp,    // [E,3]
    const float* __restrict__ denom,   // [N,3]
    float* __restrict__ out,           // [N,128] accumulator
    int n_edges)
{
    int gid  = blockIdx.x * blockDim.x + threadIdx.x;
    int e    = gid >> 5;
    int lane = gid & 31;
    if (e >= n_edges) return;

    const int s = (int)src[e];
    const int r = (int)dst[e];

    const float a0 = eexp[(size_t)e * N_HEAD + 0] /
                     fmaxf(denom[(size_t)r * N_HEAD + 0], 1e-16f);
    const float a1 = eexp[(size_t)e * N_HEAD + 1] /
                     fmaxf(denom[(size_t)r * N_HEAD + 1], 1e-16f);
    const float a2 = eexp[(size_t)e * N_HEAD + 2] /
                     fmaxf(denom[(size_t)r * N_HEAD + 2], 1e-16f);
    const float scale = (a0 + a1 + a2) * (1.0f / 3.0f);

    const float4 v =
        reinterpret_cast<const float4*>(hproj + (size_t)s * OUT_F)[lane];
    float* op = out + (size_t)r * OUT_F + lane * 4;
    atomicAdd(op + 0, v.x * scale);
    atomicAdd(op + 1, v.y * scale);
    atomicAdd(op + 2, v.z * scale);
    atomicAdd(op + 3, v.w * scale);
}

// ---------------------------------------------------------------------------
// Kernel 5: SiLU in place.
// ---------------------------------------------------------------------------
__global__ __launch_bounds__(256) void gat_silu_kernel(float* __restrict__ out,
                                                       int n) {
    int i = blockIdx.x * blockDim.x + threadIdx.x;
    if (i < n) {
        float v = out[i];
        out[i] = v / (1.0f + __expf(-v));
    }
}

// ---------------------------------------------------------------------------
// Host-side launch
// ---------------------------------------------------------------------------
extern "C" void kernel_launch(void* const* d_in, const int* in_sizes, int n_in,
                              void* d_out, int out_size, void* d_ws,
                              size_t ws_size, hipStream_t stream) {
    const float*     x   = (const float*)d_in[0];
    const long long* adj = (const long long*)d_in[1];   // int64 [2,E]
    const float*     Wq  = (const float*)d_in[2];
    const float*     bq  = (const float*)d_in[3];
    const float*     aw  = (const float*)d_in[4];
    const float*     ab  = (const float*)d_in[5];
    const float*     lw  = (const float*)d_in[6];
    const float*     lb  = (const float*)d_in[7];

    const int n_nodes = in_sizes[0] / IN_F;
    const int n_edges = in_sizes[1] / 2;

    const long long* src = adj;
    const long long* dst = adj + n_edges;

    float* out = (float*)d_out;

    // Workspace carve-up (256B aligned slabs).
    auto align_up = [](size_t v) { return (v + 255) & ~(size_t)255; };
    char*  ws  = (char*)d_ws;
    size_t off = 0;
    float* attr = (float*)(ws + off);
    off += align_up((size_t)n_nodes * ATTR_F * sizeof(float));
    float* h = (float*)(ws + off);
    off += align_up((size_t)n_nodes * OUT_F * sizeof(float));
    float* elog = (float*)(ws + off);
    off += align_up((size_t)n_edges * N_HEAD * sizeof(float));
    float* segmax = (float*)(ws + off);
    off += align_up((size_t)n_nodes * N_HEAD * sizeof(float));
    float* denom = (float*)(ws + off);
    off += align_up((size_t)n_nodes * N_HEAD * sizeof(float));
    (void)ws_size; (void)n_in;

    // 0) init accumulators
    {
        int n_seg  = n_nodes * N_HEAD;
        int n_init = out_size > n_seg ? out_size : n_seg;
        gat_init_kernel<<<(n_init + 255) / 256, 256, 0, stream>>>(
            out, out_size, segmax, denom, n_seg);
    }
    // 1) node projections (WMMA f32)
    gat_node_proj_kernel<<<(n_nodes + 15) / 16, 128, 0, stream>>>(
        x, lw, lb, Wq, bq, h, attr, n_nodes);
    // 2) edge logits + segment max
    gat_edge_logits_kernel<<<(n_edges + 255) / 256, 256, 0, stream>>>(
        src, dst, attr, aw, ab, elog, segmax, n_edges);
    // 3) exp + denominators
    gat_edge_exp_kernel<<<(n_edges + 255) / 256, 256, 0, stream>>>(
        dst, segmax, elog, denom, n_edges);
    // 4) weighted scatter (wave per edge)
    {
        long long tot = (long long)n_edges * 32;
        gat_edge_aggregate_kernel<<<(int)((tot + 255) / 256), 256, 0, stream>>>(
            src, dst, h, elog, denom, out, n_edges);
    }
    // 5) SiLU
    gat_silu_kernel<<<(out_size + 255) / 256, 256, 0, stream>>>(out, out_size);
}